// Corr2DM_20461224198795
// MI455X (gfx1250) — compile-verified
//
#include <hip/hip_runtime.h>

typedef __attribute__((ext_vector_type(2))) float v2f;
typedef __attribute__((ext_vector_type(4))) float v4f;
typedef __attribute__((ext_vector_type(8))) float v8f;

#define C_CH 64
#define DMAX 64
#define H_DIM 256
#define W_DIM 512
#define CH_STRIDE (H_DIM * W_DIM)        /* 131072 */
#define BATCH_STRIDE (128 * CH_STRIDE)

#define NB 4                 /* w-blocks per wave */
#define RS 20                /* padded LDS row stride (floats): 16B aligned, */
                             /* bank-stride 20 -> conflict-free over 16 lanes */
#define REGION (96 * RS)     /* 96 diagonal rows (d' in [-16,79]) per block */

// One wave per FOUR adjacent 16-wide w-blocks (4q..4q+3). The blocks share
// an 8-tile F2 window: per K-chunk 24 loads + 20 exact-fp32 WMMAs produce
// four 16x16x64 G-tile columns. Accumulator tiles are spilled to LDS along
// diagonals (row d' = 16p + M - N), so the banded gather
// out[d,w] = G[w, w-d]/64 is a single contiguous ds_load_b128 per 4 outputs.
//
// Out-of-range f2 columns (w-d < 0) are loaded unconditionally (addresses
// always stay inside x since f2 starts at channel 64); affected outputs are
// zeroed at the gather stage under a wave-uniform branch (only wave q=0).
__global__ __launch_bounds__(128) void corr2d_wmma_f32(
    const float* __restrict__ x, float* __restrict__ out) {
  const int tid  = threadIdx.x;
  const int wave = tid >> 5;
  const int lane = tid & 31;
  const int col  = lane & 15;   // matrix column (A: M, B: N)
  const int half = lane >> 4;   // selects K rows {0,1} vs {2,3}

  int bi = blockIdx.x;
  const int grp = bi & 1;       // 2 groups (4 waves x 64 cols) per row
  bi >>= 1;
  const int h = bi & (H_DIM - 1);
  const int b = bi >> 8;

  const int q  = grp * 4 + wave;  // 0..7
  const int w0 = q * 64;          // base column of leftmost block

  const size_t base = (size_t)b * BATCH_STRIDE + (size_t)h * W_DIM;

  // A: f1 rows c0 = 4t+2*half, c0+1; block j at column w0 + 16j + col
  const float* pa = x + base + (size_t)(2 * half) * CH_STRIDE + (w0 + col);
  // B: f2 rows; tile k has column base w0 + 48 - 16k (k = 0..7)
  const float* pb = x + base + (size_t)(C_CH + 2 * half) * CH_STRIDE + (w0 + col);

  v8f acc[NB][5] = {};   // acc[j][p] = G(block 4q+j, shifted by 16p)

#pragma unroll
  for (int t = 0; t < 16; ++t) {          // K = 64 in 16 chunks of 4
    const size_t o = (size_t)t * 4 * CH_STRIDE;
    v2f a[NB];
#pragma unroll
    for (int j = 0; j < NB; ++j) {
      a[j][0] = pa[o + 16 * j];
      a[j][1] = pa[o + CH_STRIDE + 16 * j];
    }
    v2f bm[8];                            // shared F2 window, 8 tiles
#pragma unroll
    for (int k = 0; k < 8; ++k) {
      bm[k][0] = pb[o + 48 - 16 * k];
      bm[k][1] = pb[o + CH_STRIDE + 48 - 16 * k];
    }
#pragma unroll
    for (int j = 0; j < NB; ++j) {
#pragma unroll
      for (int p = 0; p < 5; ++p) {
        acc[j][p] = __builtin_amdgcn_wmma_f32_16x16x4_f32(
            false, a[j], false, bm[3 - j + p], (short)0, acc[j][p], false, false);
      }
    }
  }

  // Diagonal spill: tile p element (M = r+8*half, N = col) -> LDS
  // [row d'+16 = 16p + M - N + 16][col M], row stride RS. One address VGPR,
  // all 160 stores use compile-time immediate offsets.
  __shared__ __align__(16) float smem[4][NB * REGION];
  float* s = &smem[wave][0];
  {
    const int sb = (8 * half - col + 16) * RS + 8 * half;  // r=0, p=0, j=0
#pragma unroll
    for (int j = 0; j < NB; ++j) {
#pragma unroll
      for (int p = 0; p < 5; ++p) {
#pragma unroll
        for (int r = 0; r < 8; ++r) {
          s[j * REGION + p * 16 * RS + sb + r * (RS + 1)] = acc[j][p][r];
        }
      }
    }
  }
  __syncthreads();

  // Gather: row d of block j is contiguous at LDS row (d+16); each lane
  // reads 4 consecutive w (ds_load_b128) and writes one global_store_b128.
  const float scale = 1.0f / 64.0f;
#pragma unroll
  for (int j = 0; j < NB; ++j) {
    const int w0b = w0 + 16 * j;
    const float* sb2 = s + j * REGION;
#pragma unroll
    for (int it = 0; it < 8; ++it) {
      const int d   = it * 8 + (lane >> 2);  // 8 d-rows per iteration
      const int wwb = (lane & 3) * 4;        // 4 consecutive w per lane
      v4f vv = *(const v4f*)(sb2 + (d + 16) * RS + wwb);
#pragma unroll
      for (int e = 0; e < 4; ++e) vv[e] *= scale;
      if (w0b < DMAX) {                      // wave-uniform: only q==0 wave
#pragma unroll
        for (int e = 0; e < 4; ++e) {
          if (w0b + wwb + e - d < 0) vv[e] = 0.0f;
        }
      }
      *(v4f*)(out + ((size_t)(b * DMAX + d) * H_DIM + h) * W_DIM + (w0b + wwb)) = vv;
    }
  }
}

extern "C" void kernel_launch(void* const* d_in, const int* in_sizes, int n_in,
                              void* d_out, int out_size, void* d_ws, size_t ws_size,
                              hipStream_t stream) {
  const float* x = (const float*)d_in[0];
  float* out = (float*)d_out;
  // grid: 4 batches * 256 rows * 2 groups (4 waves x 4 blocks x 16 cols)
  dim3 grid(4 * 256 * 2);
  dim3 block(128);
  corr2d_wmma_f32<<<grid, block, 0, stream>>>(x, out);
}